// ACFLoss_64080912056476
// MI455X (gfx1250) — compile-verified
//
#include <hip/hip_runtime.h>
#include <math.h>

typedef __attribute__((ext_vector_type(2))) float v2f;
typedef __attribute__((ext_vector_type(8))) float v8f;

#define B_DIM   256
#define S_DIM   4096
#define D_DIM   8
#define MAXLAG  64
#define KSTEPS  64            // 256 blocks of 16 / (K=4 per WMMA step)
#define PADLEN  64
#define CH_STRIDE 4177        // 4096 + 64 pad + 17 (odd -> bank spread)

// workspace (floats):
//   [0..31]    stat[input][ sum(8) | sumsq(8) ]
//   [32..1055] cacc[input][d][lag]   2*8*64
#define WS_STAT  0
#define WS_CACC  32
#define WS_TOTAL 1056

__global__ void k_zero(float* __restrict__ ws, int n) {
    int i = blockIdx.x * blockDim.x + threadIdx.x;
    if (i < n) ws[i] = 0.0f;
}

// Per-channel sum / sum-of-squares for both inputs.
__global__ void k_stats(const float* __restrict__ xf, const float* __restrict__ xr,
                        float* __restrict__ stat) {
    const int input = blockIdx.y;
    const float4* src = (const float4*)(input == 0 ? xf : xr);
    const size_t nf4 = (size_t)B_DIM * S_DIM * 2;   // two float4 per 8-ch row
    float s0=0,s1=0,s2=0,s3=0,q0=0,q1=0,q2=0,q3=0;
    const size_t stride = (size_t)gridDim.x * blockDim.x;   // even -> fi parity fixed
    for (size_t fi = (size_t)blockIdx.x * blockDim.x + threadIdx.x; fi < nf4; fi += stride) {
        float4 v = src[fi];
        s0 += v.x; s1 += v.y; s2 += v.z; s3 += v.w;
        q0 += v.x*v.x; q1 += v.y*v.y; q2 += v.z*v.z; q3 += v.w*v.w;
    }
    __shared__ float sb[16];
    if (threadIdx.x < 16) sb[threadIdx.x] = 0.0f;
    __syncthreads();
    const int c0 = (threadIdx.x & 1) * 4;
    atomicAdd(&sb[c0+0], s0); atomicAdd(&sb[c0+1], s1);
    atomicAdd(&sb[c0+2], s2); atomicAdd(&sb[c0+3], s3);
    atomicAdd(&sb[8+c0+0], q0); atomicAdd(&sb[8+c0+1], q1);
    atomicAdd(&sb[8+c0+2], q2); atomicAdd(&sb[8+c0+3], q3);
    __syncthreads();
    if (threadIdx.x < 16) atomicAdd(&stat[input*16 + threadIdx.x], sb[threadIdx.x]);
}

// One workgroup per (input, b). Wave w handles channel w.
// Gram matrices G_g[m,n] = sum_k xc[16k+m]*xc[16(k+g)+n] via V_WMMA_F32_16X16X4_F32;
// element (m,n) of G_g contributes to lag = 16g + n - m.
__global__ void k_corr(const float* __restrict__ xf, const float* __restrict__ xr,
                       const float* __restrict__ stat, float* __restrict__ cacc) {
    extern __shared__ float smem[];                 // 8*CH_STRIDE + 512 floats
    float* xs   = smem;
    float* lagb = smem + 8 * CH_STRIDE;

    const int input = blockIdx.y;
    const int b     = blockIdx.x;
    const int tid   = threadIdx.x;
    const float* src = (input == 0 ? xf : xr);
    const float invN = 1.0f / ((float)B_DIM * (float)S_DIM);

    // channel means for this thread's half-row (fi parity == tid parity)
    const int cm = (tid & 1) * 4;
    const float m0 = stat[input*16 + cm + 0] * invN;
    const float m1 = stat[input*16 + cm + 1] * invN;
    const float m2 = stat[input*16 + cm + 2] * invN;
    const float m3 = stat[input*16 + cm + 3] * invN;

    // coalesced load of the whole 8-channel slab, center, stage in LDS
    const float4* in4 = (const float4*)(src + (size_t)b * S_DIM * D_DIM);
    #pragma unroll 4
    for (int it = 0; it < 32; ++it) {
        const int fi = it * 256 + tid;
        float4 v = in4[fi];
        const int t  = fi >> 1;
        const int c0 = (fi & 1) * 4;
        xs[(c0+0)*CH_STRIDE + t] = v.x - m0;
        xs[(c0+1)*CH_STRIDE + t] = v.y - m1;
        xs[(c0+2)*CH_STRIDE + t] = v.z - m2;
        xs[(c0+3)*CH_STRIDE + t] = v.w - m3;
    }
    for (int i = tid; i < 8 * PADLEN; i += 256) {        // zero tail -> mask semantics
        xs[(i >> 6) * CH_STRIDE + S_DIM + (i & 63)] = 0.0f;
    }
    for (int i = tid; i < 8 * MAXLAG; i += 256) lagb[i] = 0.0f;
    __syncthreads();

    const int ch   = tid >> 5;          // wave id == channel
    const int lane = tid & 31;
    const int l15  = lane & 15;
    const int hi   = lane >> 4;         // lanes 16-31 carry K=2,3 (and M=r+8 rows of C)
    const float* xp = xs + ch * CH_STRIDE;

    v8f acc[5];
    #pragma unroll
    for (int g = 0; g < 5; ++g) { v8f z = {}; acc[g] = z; }

    for (int s = 0; s < KSTEPS; ++s) {
        const int base = s * 64 + hi * 32 + l15;
        const float L0 = xp[base +  0];
        const float L1 = xp[base + 16];
        const float L2 = xp[base + 32];
        const float L3 = xp[base + 48];
        const float L4 = xp[base + 64];
        const float L5 = xp[base + 80];
        v2f A  = {L0, L1};
        v2f B0 = {L0, L1}; v2f B1 = {L1, L2}; v2f B2 = {L2, L3};
        v2f B3 = {L3, L4}; v2f B4 = {L4, L5};
        acc[0] = __builtin_amdgcn_wmma_f32_16x16x4_f32(false, A, false, B0, (short)0, acc[0], false, false);
        acc[1] = __builtin_amdgcn_wmma_f32_16x16x4_f32(false, A, false, B1, (short)0, acc[1], false, false);
        acc[2] = __builtin_amdgcn_wmma_f32_16x16x4_f32(false, A, false, B2, (short)0, acc[2], false, false);
        acc[3] = __builtin_amdgcn_wmma_f32_16x16x4_f32(false, A, false, B3, (short)0, acc[3], false, false);
        acc[4] = __builtin_amdgcn_wmma_f32_16x16x4_f32(false, A, false, B4, (short)0, acc[4], false, false);
    }

    // scatter Gram elements into per-channel lag bins
    #pragma unroll
    for (int g = 0; g < 5; ++g) {
        #pragma unroll
        for (int r = 0; r < 8; ++r) {
            const int M   = r + hi * 8;
            const int lag = 16 * g + l15 - M;
            if (lag >= 0 && lag < MAXLAG)
                atomicAdd(&lagb[ch * MAXLAG + lag], acc[g][r]);
        }
    }
    __syncthreads();

    for (int i = tid; i < 8 * MAXLAG; i += 256)
        atomicAdd(&cacc[(size_t)input * 8 * MAXLAG + i], lagb[i]);
}

__global__ void k_final(const float* __restrict__ stat, const float* __restrict__ cacc,
                        float* __restrict__ out) {
    __shared__ float sums[8];
    __shared__ float total;
    const int tid = threadIdx.x;            // 512 threads: (d, lag)
    if (tid < 8)  sums[tid] = 0.0f;
    if (tid == 0) total = 0.0f;
    __syncthreads();
    const int d   = tid >> 6;
    const int lag = tid & 63;
    const float N = (float)B_DIM * (float)S_DIM;
    float acf[2];
    #pragma unroll
    for (int i = 0; i < 2; ++i) {
        const float mean = stat[i*16 + d] / N;
        const float var  = stat[i*16 + 8 + d] / N - mean * mean;
        const float c    = cacc[i * 8 * MAXLAG + d * MAXLAG + lag];
        acf[i] = c / ((float)B_DIM * (float)(S_DIM - lag)) / var;
    }
    const float diff = acf[0] - acf[1];     // fake - real
    atomicAdd(&sums[d], diff * diff);
    __syncthreads();
    if (lag == 0) atomicAdd(&total, sqrtf(sums[d]));
    __syncthreads();
    if (tid == 0) out[0] = total * (1.0f / 8.0f);   // REG * mean over channels
}

extern "C" void kernel_launch(void* const* d_in, const int* in_sizes, int n_in,
                              void* d_out, int out_size, void* d_ws, size_t ws_size,
                              hipStream_t stream) {
    const float* xf = (const float*)d_in[0];   // x_fake
    const float* xr = (const float*)d_in[1];   // x_real
    float* ws   = (float*)d_ws;
    float* stat = ws + WS_STAT;
    float* cacc = ws + WS_CACC;
    float* out  = (float*)d_out;

    k_zero<<<dim3((WS_TOTAL + 255) / 256), dim3(256), 0, stream>>>(ws, WS_TOTAL);
    k_stats<<<dim3(512, 2), dim3(256), 0, stream>>>(xf, xr, stat);
    const size_t smem = (size_t)(8 * CH_STRIDE + 8 * MAXLAG) * sizeof(float); // ~135 KB, legal in 320 KB WGP LDS
    k_corr<<<dim3(B_DIM, 2), dim3(256), smem, stream>>>(xf, xr, stat, cacc);
    k_final<<<dim3(1), dim3(512), 0, stream>>>(stat, cacc, out);
}